// ConstraintWholePoseScoringModule_40054865002541
// MI455X (gfx1250) — compile-verified
//
#include <hip/hip_runtime.h>

// CDNA5 / gfx1250, wave32.
typedef __attribute__((ext_vector_type(2))) float v2f;
typedef __attribute__((ext_vector_type(8))) float v8f;

#define NPOSES 16
#define NBLOCKS 1024
#define ATOMS_PER_BLOCK 16
#define NATOMS (NBLOCKS * ATOMS_PER_BLOCK)

// Whole problem = 3 constraints (compile-time constants in the reference),
// ~30 FLOPs, ~100 bytes of memory traffic -> single wave32, single workgroup,
// branch-free, no atomics (deterministic). Squared distances are computed as
// the diagonal of a Gram matrix via V_WMMA_F32_16X16X4_F32:
//   A row m = diff_m = (dx,dy,dz,0), B col n = diff_n  =>  D[m][m] = |diff_m|^2
__global__ __launch_bounds__(32)
void ConstraintWholePoseScoringModule_40054865002541_kernel(
    const float* __restrict__ coords,        // (NPOSES, NATOMS, 3) f32
    const int*   __restrict__ bco,           // (NPOSES, NBLOCKS) i32
    float*       __restrict__ out)           // (1, NPOSES) f32
{
    const int lane = threadIdx.x;            // 0..31, exactly one wave
    const int M    = lane & 15;              // constraint row in the 16x4 A tile

    // Baked-in constraint table (module constants in the reference):
    //   c0: (res 0, atom 0) <-> (res 1, atom 1), pose 0
    //   c1: (res 1, atom 0) <-> (res 2, atom 1), pose 0
    //   c2: (res 0, atom 0) <-> (res 1, atom 1), pose 0
    const bool valid = (M < 3);
    const int  cm    = valid ? M : 0;        // clamp for safe addressing
    const int  res0  = (cm == 1) ? 1 : 0;
    const int  res1  = (cm == 1) ? 2 : 1;
    const int  atom0 = 0;
    const int  atom1 = 1;
    const int  pose  = 0;

    // gidx = block_coord_offset[pose, res] + atom
    const int g0 = bco[pose * NBLOCKS + res0] + atom0;
    const int g1 = bco[pose * NBLOCKS + res1] + atom1;

    const float* p0 = coords + ((size_t)pose * NATOMS + (size_t)g0) * 3;
    const float* p1 = coords + ((size_t)pose * NATOMS + (size_t)g1) * 3;

    float dx = p0[0] - p1[0];
    float dy = p0[1] - p1[1];
    float dz = p0[2] - p1[2];
    // Zero the 13 padding rows so the Gram diagonal stays finite.
    dx = valid ? dx : 0.0f;
    dy = valid ? dy : 0.0f;
    dz = valid ? dz : 0.0f;

    // 32-bit A (16x4) VGPR layout: lanes 0-15 carry K=0,1; lanes 16-31 carry
    // K=2,3 (K=3 is zero padding). B (4x16) mirrors this, so A and B use the
    // same per-lane packing when B's columns equal A's rows.
    v2f ab;
    ab.x = (lane < 16) ? dx : dz;
    ab.y = (lane < 16) ? dy : 0.0f;

    v8f acc = {};
    // 8 args: (neg_a, A, neg_b, B, c_mod, C, reuse_a, reuse_b)
    acc = __builtin_amdgcn_wmma_f32_16x16x4_f32(
        /*neg_a=*/false, ab, /*neg_b=*/false, ab,
        /*c_mod=*/(short)0, acc, /*reuse_a=*/false, /*reuse_b=*/false);

    // C/D 16x16 f32 layout: D[m][m] lives in VGPR m at lane m (m < 8).
    // Lanes 0..2 therefore hold the three squared distances.
    const float ss = (lane == 0) ? acc[0] : ((lane == 1) ? acc[1] : acc[2]);

    float d     = __builtin_sqrtf(ss) - 4.0f;   // IDEAL = 4.0
    float score = d * d;

    // Deterministic fixed-order reduction: broadcast the three scores.
    const float s0 = __shfl(score, 0, 32);
    const float s1 = __shfl(score, 1, 32);
    const float s2 = __shfl(score, 2, 32);

    // out[p] = sum of scores for constraints on pose p; all constraints are
    // on pose 0. Write every output element (d_out is poisoned by harness).
    if (lane < NPOSES) {
        out[lane] = (lane == 0) ? (s0 + s1 + s2) : 0.0f;
    }
}

extern "C" void kernel_launch(void* const* d_in, const int* in_sizes, int n_in,
                              void* d_out, int out_size, void* d_ws, size_t ws_size,
                              hipStream_t stream) {
    const float* coords = (const float*)d_in[0];   // (16, 16384, 3) f32
    const int*   bco    = (const int*)d_in[1];     // (16, 1024) i32
    float*       out    = (float*)d_out;           // 16 f32

    (void)in_sizes; (void)n_in; (void)out_size; (void)d_ws; (void)ws_size;

    ConstraintWholePoseScoringModule_40054865002541_kernel<<<1, 32, 0, stream>>>(
        coords, bco, out);
}